// hybridDynamicSimNew_51505247814195
// MI455X (gfx1250) — compile-verified
//
#include <hip/hip_runtime.h>
#include <hip/hip_bf16.h>
#include <math.h>

typedef float v2f __attribute__((ext_vector_type(2)));
typedef float v8f __attribute__((ext_vector_type(8)));

#define B_TOT 65536
#define F_STEPS 50
#define DT_C 0.01f
#define PI_F 3.14159265358979323846f

// cross-half-wave swap (lane L <-> lane L^16), group-of-32 ds_swizzle:
// offset[14:10]=xor=0x10, or=0, and=0x1f  -> 0x401f
__device__ __forceinline__ float swz16(float x) {
  return __int_as_float(__builtin_amdgcn_ds_swizzle(__float_as_int(x), 0x401f));
}

// gfx1250 hardware tanh (TRANS32 pipe) instead of the __ocml polynomial.
__device__ __forceinline__ float fast_tanh(float x) {
#if __has_builtin(__builtin_amdgcn_tanhf)
  return __builtin_amdgcn_tanhf(x);
#else
  float r;
  asm("v_tanh_f32 %0, %1" : "=v"(r) : "v"(x));
  return r;
#endif
}

__global__ __launch_bounds__(256) void hybrid_sim_kernel(
    const float* __restrict__ full_states,
    const float* __restrict__ actions,
    const float* __restrict__ fc1_w,
    const float* __restrict__ fc1_b,
    const float* __restrict__ fc2_w,
    const float* __restrict__ fc2_b,
    const float* __restrict__ params,
    const int*   __restrict__ enable_rnn,
    float* __restrict__ out)
{
  const int tid  = blockIdx.x * blockDim.x + threadIdx.x;   // element id
  const int lane = threadIdx.x & 31;
  const bool lo  = (lane < 16);
  const int row  = lane & 15;
  const int er   = enable_rnn[0];

  // uniform params
  const float p_lf  = params[0], p_lr  = params[1], p_mIz = params[2];
  const float p_cm1 = params[3], p_cm2 = params[4], p_cr  = params[5], p_cd = params[6];
  const float p_tB  = params[7], p_tC  = params[8], p_tD  = params[9];

  // ---- fc1 weights into WMMA A-layout: M=16 outputs, K=28 (7 chunks of 4) ----
  // A VGPR0: lanes0-15 -> W[M=lane][K=4k], lanes16-31 -> W[M=lane-16][K=4k+2]
  v2f A1[7];
#pragma unroll
  for (int k = 0; k < 7; ++k) {
    int c0 = 4 * k + (lo ? 0 : 2);
    A1[k].x = (c0     < 25) ? fc1_w[row * 25 + c0]     : 0.0f;
    A1[k].y = (c0 + 1 < 25) ? fc1_w[row * 25 + c0 + 1] : 0.0f;
  }
  // ---- fc2 weights: M=16 (rows 3..15 zero), K=16 (4 chunks) ----
  v2f A2[4];
#pragma unroll
  for (int k = 0; k < 4; ++k) {
    int c0 = 4 * k + (lo ? 0 : 2);
    A2[k].x = (row < 3) ? fc2_w[row * 16 + c0]     : 0.0f;
    A2[k].y = (row < 3) ? fc2_w[row * 16 + c0 + 1] : 0.0f;
  }
  // ---- biases in C/D layout: VGPR m at lane L = b[m + (L<16?0:8)] ----
  v8f C1, C2;
#pragma unroll
  for (int m = 0; m < 8; ++m) {
    int mm = m + (lo ? 0 : 8);
    C1[m] = fc1_b[mm];
    C2[m] = (mm < 3) ? fc2_b[mm] : 0.0f;
  }

  // ---- load history window (5 x 8) into registers ----
  float w[5][8];
#pragma unroll
  for (int h = 0; h < 5; ++h) {
    float4 p0 = *(const float4*)(full_states + (size_t)tid * 40 + h * 8);
    float4 p1 = *(const float4*)(full_states + (size_t)tid * 40 + h * 8 + 4);
    w[h][0] = p0.x; w[h][1] = p0.y; w[h][2] = p0.z; w[h][3] = p0.w;
    w[h][4] = p1.x; w[h][5] = p1.y; w[h][6] = p1.z; w[h][7] = p1.w;
  }

  const float* act_ptr = actions + (size_t)tid * (F_STEPS * 2);
  float*       out_ptr = out     + (size_t)tid * (F_STEPS * 8);

  for (int f = 0; f < F_STEPS; ++f) {
    // ---- features from history window ----
    float feat[28];
    float c4 = 0.0f, s4 = 0.0f;
#pragma unroll
    for (int h = 0; h < 5; ++h) {
      float ph = w[h][4];
      float ch = __cosf(ph), sh = __sinf(ph);
      feat[h]      = w[h][1] * ch + w[h][3] * sh;   // Vx history
      feat[5 + h]  = w[h][3] * ch - w[h][1] * sh;   // Vy history
      feat[10 + h] = w[h][5];
      feat[15 + h] = w[h][6];
      feat[20 + h] = w[h][7];
      if (h == 4) { c4 = ch; s4 = sh; }
    }
    feat[25] = 0.0f; feat[26] = 0.0f; feat[27] = 0.0f;

    // ---- physics on latest state ----
    float thr = w[4][6], steer = w[4][7], psi = w[4][4];
    float Vx = feat[4], Vy = feat[9];
    float t0 = atanf(Vy / Vx);
    float slip_f = steer - t0;
    float slip_r = -t0;
    float laf = p_tD * __sinf(p_tC * atanf(p_tB * slip_f));
    float lar = p_tD * __sinf(p_tC * atanf(p_tB * slip_r));
    float cs = __cosf(steer), ss = __sinf(steer);
    float fwd = (p_cm1 - p_cm2 * Vx) * thr - p_cr - p_cd * Vx * Vx;
    Vx = Vx + (fwd - laf * ss) * DT_C;
    Vy = Vy + (lar + laf * cs) * DT_C;
    float omega = p_mIz * (laf * p_lf * cs - lar * p_lr) * DT_C;

    // ---- MLP: D(16x16) = W(16xK) x feat(Kx16 elems) via V_WMMA_F32_16X16X4_F32 ----
    if (er) {
      v8f D0 = C1, D1 = C1;        // tile0: elems lanes0-15, tile1: elems lanes16-31
#pragma unroll
      for (int k = 0; k < 7; ++k) {
        float f0 = feat[4 * k],     f1 = feat[4 * k + 1];
        float f2 = feat[4 * k + 2], f3 = feat[4 * k + 3];
        float s0 = swz16(f0), s1 = swz16(f1), s2 = swz16(f2), s3 = swz16(f3);
        v2f b0, b1;
        b0.x = lo ? f0 : s2;  b0.y = lo ? f1 : s3;   // B for tile0
        b1.x = lo ? s0 : f2;  b1.y = lo ? s1 : f3;   // B for tile1
        D0 = __builtin_amdgcn_wmma_f32_16x16x4_f32(false, A1[k], false, b0, (short)0, D0, false, false);
        D1 = __builtin_amdgcn_wmma_f32_16x16x4_f32(false, A1[k], false, b1, (short)0, D1, false, false);
      }
      // tanh activation (elementwise, layout-agnostic) on hardware TRANS pipe
      v8f Y0, Y1;
#pragma unroll
      for (int m = 0; m < 8; ++m) { Y0[m] = fast_tanh(D0[m]); Y1[m] = fast_tanh(D1[m]); }

      // fc2: reshape Y (D-layout) into B-layout chunks, K=16 in 4 chunks
      float z0[8], z1[8];
#pragma unroll
      for (int m = 0; m < 8; ++m) { z0[m] = swz16(Y0[m]); z1[m] = swz16(Y1[m]); }

      v8f E0 = C2, E1 = C2;
#pragma unroll
      for (int k = 0; k < 4; ++k) {
        int base = (k & 1) * 4;
        v2f b0, b1;
        if (k < 2) {  // K rows 0..7 live in lanes0-15 of Y
          b0.x = lo ? Y0[base]     : z0[base + 2];
          b0.y = lo ? Y0[base + 1] : z0[base + 3];
          b1.x = lo ? Y1[base]     : z1[base + 2];
          b1.y = lo ? Y1[base + 1] : z1[base + 3];
        } else {      // K rows 8..15 live in lanes16-31 of Y
          b0.x = lo ? z0[base]     : Y0[base + 2];
          b0.y = lo ? z0[base + 1] : Y0[base + 3];
          b1.x = lo ? z1[base]     : Y1[base + 2];
          b1.y = lo ? z1[base + 1] : Y1[base + 3];
        }
        E0 = __builtin_amdgcn_wmma_f32_16x16x4_f32(false, A2[k], false, b0, (short)0, E0, false, false);
        E1 = __builtin_amdgcn_wmma_f32_16x16x4_f32(false, A2[k], false, b1, (short)0, E1, false, false);
      }
      // scatter y2 rows 0..2 back to owning lanes: tile1 results sit in lanes0-15
      float g0 = swz16(E1[0]), g1 = swz16(E1[1]), g2 = swz16(E1[2]);
      float r0 = lo ? E0[0] : g0;
      float r1 = lo ? E0[1] : g1;
      float r2 = lo ? E0[2] : g2;
      Vx    += fast_tanh(r0) * 5.0f * DT_C;
      Vy    += fast_tanh(r1) * 5.0f * DT_C;
      omega += fast_tanh(r2) * 3.0f * DT_C;
    }

    // ---- rotate to global frame, wrap heading ----
    float Vxg = Vx * c4 - Vy * s4;
    float Vyg = Vx * s4 + Vy * c4;
    float ang = psi + omega * DT_C + PI_F;
    float q = ang * (1.0f / (2.0f * PI_F));
    q = q - floorf(q);
    float psi_new = q * (2.0f * PI_F) - PI_F;

    float2 av = *(const float2*)(act_ptr + f * 2);
    float nf[8];
    nf[0] = w[4][0] + Vxg * DT_C;
    nf[1] = Vxg;
    nf[2] = w[4][2] + Vyg * DT_C;
    nf[3] = Vyg;
    nf[4] = psi_new;
    nf[5] = omega;
    nf[6] = av.x;
    nf[7] = av.y;

    float4 o0 = { nf[0], nf[1], nf[2], nf[3] };
    float4 o1 = { nf[4], nf[5], nf[6], nf[7] };
    *(float4*)(out_ptr + f * 8)     = o0;
    *(float4*)(out_ptr + f * 8 + 4) = o1;

    // ---- shift window ----
#pragma unroll
    for (int h = 0; h < 4; ++h)
#pragma unroll
      for (int i = 0; i < 8; ++i) w[h][i] = w[h + 1][i];
#pragma unroll
    for (int i = 0; i < 8; ++i) w[4][i] = nf[i];
  }
}

extern "C" void kernel_launch(void* const* d_in, const int* in_sizes, int n_in,
                              void* d_out, int out_size, void* d_ws, size_t ws_size,
                              hipStream_t stream) {
  (void)in_sizes; (void)n_in; (void)out_size; (void)d_ws; (void)ws_size;
  const float* full_states = (const float*)d_in[0];
  const float* actions     = (const float*)d_in[1];
  const float* fc1_w       = (const float*)d_in[2];
  const float* fc1_b       = (const float*)d_in[3];
  const float* fc2_w       = (const float*)d_in[4];
  const float* fc2_b       = (const float*)d_in[5];
  const float* params      = (const float*)d_in[6];
  const int*   er          = (const int*)d_in[7];
  float* out = (float*)d_out;

  dim3 grid(B_TOT / 256), block(256);
  hipLaunchKernelGGL(hybrid_sim_kernel, grid, block, 0, stream,
                     full_states, actions, fc1_w, fc1_b, fc2_w, fc2_b, params, er, out);
}